// Block_Height_Reducing_Filtering_47158740910393
// MI455X (gfx1250) — compile-verified
//
#include <hip/hip_runtime.h>
#include <hip/hip_bf16.h>

#define NH 4
#define BB 16
#define CC 128
#define HH 2000
#define DD 500
#define DP 512
#define CO 128
#define KNC (NH*CC)          // 512
#define NCH (HH/16)          // 125 h-chunks
#define XPAD 20              // padded LDS row (floats): 80B, 16B-aligned, conflict-free
#define BN_EPS 1e-5f

typedef __attribute__((ext_vector_type(2))) float v2f;
typedef __attribute__((ext_vector_type(8))) float v8f;

// D = A(16x4) * B(4x16) + C, fp32 WMMA (CDNA5, wave32)
__device__ __forceinline__ v8f wmma_f32(v2f a, v2f b, v8f c) {
    return __builtin_amdgcn_wmma_f32_16x16x4_f32(false, a, false, b, (short)0, c, false, false);
}

// Async-copy one x chunk (128 rows x 16 floats) into LDS [row][XPAD].
// 16 x global_load_async_to_lds_b128: lane -> row = i*8 + lane/4, 16B sub-chunk = lane%4.
__device__ __forceinline__ void issue_x_chunk(const float* __restrict__ gbase,
                                              uint32_t lds_base, int lane) {
    const int sub = lane & 3;
    const int r0  = lane >> 2;
    const float* g = gbase + (size_t)r0 * HH + sub * 4;
    uint32_t l = lds_base + (uint32_t)(r0 * (XPAD * 4) + sub * 16);
#pragma unroll
    for (int i = 0; i < 16; ++i) {
        asm volatile("global_load_async_to_lds_b128 %0, %1, off"
                     :: "v"(l), "v"(g) : "memory");
        g += 8 * HH;
        l += 8 * (XPAD * 4);
    }
}
#define ASYNC_WAIT_LE16() asm volatile("s_wait_asynccnt 0x10" ::: "memory")
#define ASYNC_WAIT_0()    asm volatile("s_wait_asynccnt 0x0"  ::: "memory")

// ---------------------------------------------------------------------------
// K1: softmax row stats (max, sum(exp)) per (n,b,d).
// 1 wave per (n, b, d-tile of 16). x chunks double-buffered via async->LDS.
// Per-lane online (m,l); single cross-lane merge at the end.
// ---------------------------------------------------------------------------
__global__ void k1_softmax_stats(const float* __restrict__ x,
                                 const float* __restrict__ Wpool,
                                 float* __restrict__ rowmax,
                                 float* __restrict__ rowsum) {
    const int wg = blockIdx.x;
    const int dt = wg & 31;
    const int b  = (wg >> 5) & 15;
    const int n  = wg >> 9;
    const int lane = threadIdx.x;           // blockDim.x == 32
    const int hi = lane >> 4, nl = lane & 15;

    __shared__ __attribute__((aligned(16))) float Xs[2][CC * XPAD];  // 2 x 10KB
    __shared__ float Wt[16 * CC];                                    // 8KB
    for (int i = lane; i < 16 * CC; i += 32) {
        int r = i / CC, col = i % CC;
        int d = dt * 16 + r;
        Wt[i] = (d < DD) ? Wpool[((size_t)n * DD + d) * CC + col] : 0.0f;
    }
    __syncthreads();

    const float* xb = x + (size_t)b * CC * HH;
    const uint32_t xs0 = (uint32_t)(uintptr_t)&Xs[0][0];
    const uint32_t xs1 = (uint32_t)(uintptr_t)&Xs[1][0];

    issue_x_chunk(xb, xs0, lane);           // prefetch chunk 0

    float m[8], l[8];
#pragma unroll
    for (int v = 0; v < 8; ++v) { m[v] = -3.0e38f; l[v] = 0.0f; }

    for (int hc = 0; hc < NCH; ++hc) {
        if (hc + 1 < NCH) {
            issue_x_chunk(xb + (size_t)(hc + 1) * 16, ((hc + 1) & 1) ? xs1 : xs0, lane);
            ASYNC_WAIT_LE16();              // chunk hc is resident
        } else {
            ASYNC_WAIT_0();
        }
        const float* X = Xs[hc & 1];
        v8f s{};
#pragma unroll 8
        for (int k0 = 0; k0 < CC; k0 += 4) {
            const int kb = k0 + (hi << 1);
            v2f a;  a.x = Wt[nl * CC + kb];       a.y = Wt[nl * CC + kb + 1];
            v2f bf; bf.x = X[kb * XPAD + nl];     bf.y = X[(kb + 1) * XPAD + nl];
            s = wmma_f32(a, bf, s);
        }
        // per-lane online softmax update (lane owns column h = nl of rows v+8*hi)
#pragma unroll
        for (int v = 0; v < 8; ++v) {
            float sv = s[v];
            float mn = fmaxf(m[v], sv);
            l[v] = l[v] * __expf(m[v] - mn) + __expf(sv - mn);
            m[v] = mn;
        }
    }
    // merge the 16 column-owners (xor masks < 16 stay within each half-wave)
#pragma unroll
    for (int v = 0; v < 8; ++v) {
#pragma unroll
        for (int off = 1; off <= 8; off <<= 1) {
            float mo = __shfl_xor(m[v], off, 32);
            float lo = __shfl_xor(l[v], off, 32);
            float mn = fmaxf(m[v], mo);
            l[v] = l[v] * __expf(m[v] - mn) + lo * __expf(mo - mn);
            m[v] = mn;
        }
    }
    if (nl == 0) {
#pragma unroll
        for (int v = 0; v < 8; ++v) {
            int d = dt * 16 + v + 8 * hi;
            size_t idx = ((size_t)(n * BB + b)) * DP + d;
            rowmax[idx] = m[v];
            rowsum[idx] = l[v];
        }
    }
}

// ---------------------------------------------------------------------------
// K2: recompute s, normalize to P in LDS, pooling GEMM ht = x @ P^T.
// x chunks double-buffered via async->LDS; chunk feeds BOTH GEMMs from LDS.
// Output ht[n][b][d(DP)][c], padded d-rows exactly zero.
// ---------------------------------------------------------------------------
__global__ void k2_pool_apply(const float* __restrict__ x,
                              const float* __restrict__ Wpool,
                              const float* __restrict__ rowmax,
                              const float* __restrict__ rowsum,
                              float* __restrict__ ht) {
    const int wg = blockIdx.x;
    const int dt = wg & 31;
    const int b  = (wg >> 5) & 15;
    const int n  = wg >> 9;
    const int lane = threadIdx.x;
    const int hi = lane >> 4, nl = lane & 15;

    __shared__ __attribute__((aligned(16))) float Xs[2][CC * XPAD];
    __shared__ float Wt[16 * CC];
    __shared__ float Pt[16 * 16];           // P^T chunk: [h_local][d_local]
    for (int i = lane; i < 16 * CC; i += 32) {
        int r = i / CC, col = i % CC;
        int d = dt * 16 + r;
        Wt[i] = (d < DD) ? Wpool[((size_t)n * DD + d) * CC + col] : 0.0f;
    }
    float rm[8], rinv[8];
#pragma unroll
    for (int v = 0; v < 8; ++v) {
        int d = dt * 16 + v + 8 * hi;
        size_t idx = ((size_t)(n * BB + b)) * DP + d;
        float mm = rowmax[idx], ll = rowsum[idx];
        rm[v]   = mm;
        rinv[v] = (d < DD && ll > 0.0f) ? 1.0f / ll : 0.0f;   // zero padded rows
    }
    __syncthreads();

    const float* xb = x + (size_t)b * CC * HH;
    const uint32_t xs0 = (uint32_t)(uintptr_t)&Xs[0][0];
    const uint32_t xs1 = (uint32_t)(uintptr_t)&Xs[1][0];

    issue_x_chunk(xb, xs0, lane);

    v8f acc[8] = {};                        // 8 c-tiles x (16c x 16d)

    for (int hc = 0; hc < NCH; ++hc) {
        if (hc + 1 < NCH) {
            issue_x_chunk(xb + (size_t)(hc + 1) * 16, ((hc + 1) & 1) ? xs1 : xs0, lane);
            ASYNC_WAIT_LE16();
        } else {
            ASYNC_WAIT_0();
        }
        const float* X = Xs[hc & 1];
        v8f s{};
#pragma unroll 8
        for (int k0 = 0; k0 < CC; k0 += 4) {
            const int kb = k0 + (hi << 1);
            v2f a;  a.x = Wt[nl * CC + kb];       a.y = Wt[nl * CC + kb + 1];
            v2f bf; bf.x = X[kb * XPAD + nl];     bf.y = X[(kb + 1) * XPAD + nl];
            s = wmma_f32(a, bf, s);
        }
        __syncthreads();
#pragma unroll
        for (int v = 0; v < 8; ++v) {
            float p = __expf(s[v] - rm[v]) * rinv[v];   // softmax prob (0 on pads)
            Pt[nl * 16 + v + 8 * hi] = p;               // Pt[h][d]
        }
        __syncthreads();
        // ht_tile += x_chunk (128 x 16h) @ Pt (16h x 16d), both operands in LDS
#pragma unroll
        for (int k0 = 0; k0 < 16; k0 += 4) {
            const int kb = k0 + (hi << 1);
            v2f bf; bf.x = Pt[kb * 16 + nl]; bf.y = Pt[(kb + 1) * 16 + nl];
#pragma unroll
            for (int t = 0; t < 8; ++t) {
                const int cM = t * 16 + nl;
                v2f a; a.x = X[cM * XPAD + kb]; a.y = X[cM * XPAD + kb + 1];
                acc[t] = wmma_f32(a, bf, acc[t]);
            }
        }
    }
    // D frag: M = c = t*16 + v + 8*hi ; N = d = dt*16 + nl
    const int d = dt * 16 + nl;
    float* htb = ht + ((size_t)(n * BB + b)) * DP * CC;
#pragma unroll
    for (int t = 0; t < 8; ++t)
#pragma unroll
        for (int v = 0; v < 8; ++v) {
            int c = t * 16 + v + 8 * hi;
            htb[(size_t)d * CC + c] = acc[t][v];
        }
}

// ---------------------------------------------------------------------------
// K3: pointwise conv  y[n,b,e,c] = sum_d W_pw[n,e,d] * ht[n,b,d,c]  (WMMA)
// 1 wave per (n, b, e-tile).  W_pw e-tile staged zero-padded to K=512 in LDS.
// ---------------------------------------------------------------------------
__global__ void k3_pw(const float* __restrict__ Wpw,
                      const float* __restrict__ ht,
                      float* __restrict__ y) {
    const int wg = blockIdx.x;
    const int et = wg & 31;
    const int b  = (wg >> 5) & 15;
    const int n  = wg >> 9;
    const int lane = threadIdx.x;
    const int hi = lane >> 4, nl = lane & 15;

    __shared__ float Wt[16 * DP];            // 32 KB, zero-padded
    for (int i = lane; i < 16 * DP; i += 32) {
        int r = i / DP, col = i % DP;
        int e = et * 16 + r;
        Wt[i] = (e < DD && col < DD) ? Wpw[((size_t)n * DD + e) * DD + col] : 0.0f;
    }
    __syncthreads();

    const float* htb = ht + ((size_t)(n * BB + b)) * DP * CC;
    v8f acc[8] = {};
    for (int k0 = 0; k0 < DP; k0 += 4) {
        const int kb = k0 + (hi << 1);
        v2f a; a.x = Wt[nl * DP + kb]; a.y = Wt[nl * DP + kb + 1];
#pragma unroll
        for (int t = 0; t < 8; ++t) {
            const int cN = t * 16 + nl;
            v2f bf; bf.x = htb[(size_t)kb * CC + cN];
                    bf.y = htb[(size_t)(kb + 1) * CC + cN];
            acc[t] = wmma_f32(a, bf, acc[t]);
        }
    }
    float* yb = y + ((size_t)(n * BB + b)) * DP * CC;
#pragma unroll
    for (int t = 0; t < 8; ++t)
#pragma unroll
        for (int v = 0; v < 8; ++v) {
            int e = et * 16 + v + 8 * hi;
            int c = t * 16 + nl;
            yb[(size_t)e * CC + c] = acc[t][v];      // padded e-rows are zeros
        }
}

// ---------------------------------------------------------------------------
// K4: BN stats for pointwise branch: per (n,e) over (b,c)  (biased variance)
// ---------------------------------------------------------------------------
__global__ void k4_bnstats_pw(const float* __restrict__ y,
                              float* __restrict__ mean, float* __restrict__ var) {
    const int n = blockIdx.x / DP;
    const int e = blockIdx.x % DP;
    const int tid = threadIdx.x;
    float s = 0.0f, s2 = 0.0f;
    for (int i = tid; i < BB * CC; i += blockDim.x) {
        int b = i / CC, c = i % CC;
        float v = y[(((size_t)(n * BB + b)) * DP + e) * CC + c];
        s += v; s2 += v * v;
    }
    __shared__ float ss[256], ss2[256];
    ss[tid] = s; ss2[tid] = s2;
    __syncthreads();
    for (int off = 128; off > 0; off >>= 1) {
        if (tid < off) { ss[tid] += ss[tid + off]; ss2[tid] += ss2[tid + off]; }
        __syncthreads();
    }
    if (tid == 0) {
        const float invN = 1.0f / (float)(BB * CC);
        float mu = ss[0] * invN;
        mean[blockIdx.x] = mu;
        var[blockIdx.x]  = ss2[0] * invN - mu * mu;
    }
}

// ---------------------------------------------------------------------------
// K5: z = ht + relu(bn(y)), in-place over y.  Padded e-rows forced to 0.
// ---------------------------------------------------------------------------
__global__ void k5_apply(const float* __restrict__ ht,
                         const float* __restrict__ mean, const float* __restrict__ var,
                         const float* __restrict__ g, const float* __restrict__ beta,
                         float* __restrict__ z) {
    size_t i = (size_t)blockIdx.x * blockDim.x + threadIdx.x;
    const size_t total = (size_t)NH * BB * DP * CC;
    if (i >= total) return;
    size_t r = i / CC;
    int e = (int)(r % DP);
    size_t r2 = r / DP;
    int n = (int)(r2 / BB);
    if (e >= DD) { z[i] = 0.0f; return; }
    float yy = z[i];                                   // z aliases y
    float mu = mean[n * DP + e], vv = var[n * DP + e];
    float t = (yy - mu) * rsqrtf(vv + BN_EPS) * g[n * DD + e] + beta[n * DD + e];
    z[i] = ht[i] + fmaxf(t, 0.0f);
}

// ---------------------------------------------------------------------------
// K6: merge GEMM  m[b,o,dd] = sum_{k=n*C+c} W_merge[o,k] * z[n,b,dd,c]
// One 256-thread WG per (b, dd-tile): z panel (512 x 16) loaded coalesced
// into LDS (pad-17, conflict-free); 8 waves each run one o-tile K-loop.
// ---------------------------------------------------------------------------
#define ZPAD 17
__global__ void k6_merge(const float* __restrict__ Wm,
                         const float* __restrict__ z,
                         float* __restrict__ mbuf) {
    const int dt = blockIdx.x & 31;
    const int b  = blockIdx.x >> 5;
    const int tid = threadIdx.x;
    const int lane = tid & 31;
    const int ot = tid >> 5;                 // wave id = o-tile
    const int hi = lane >> 4, nl = lane & 15;

    __shared__ float Lz[KNC * ZPAD];         // ~34.8 KB, layout [k][dd]
    for (int i = tid; i < KNC * 16; i += 256) {
        int c  = i & 127;
        int n  = (i >> 7) & 3;
        int dd = i >> 9;
        float v = z[(((size_t)(n * BB + b)) * DP + dt * 16 + dd) * CC + c];
        Lz[(n * CC + c) * ZPAD + dd] = v;    // transpose to [k][dd]
    }
    __syncthreads();

    v8f acc{};
    for (int k0 = 0; k0 < KNC; k0 += 4) {
        const int kb = k0 + (hi << 1);
        const int oM = ot * 16 + nl;
        v2f a;  a.x = Wm[(size_t)oM * KNC + kb]; a.y = Wm[(size_t)oM * KNC + kb + 1];
        v2f bf; bf.x = Lz[kb * ZPAD + nl];       bf.y = Lz[(kb + 1) * ZPAD + nl];
        acc = wmma_f32(a, bf, acc);
    }
    const int ddN = dt * 16 + nl;
#pragma unroll
    for (int v = 0; v < 8; ++v) {
        int o = ot * 16 + v + 8 * hi;
        mbuf[((size_t)b * CO + o) * DP + ddN] = acc[v];  // padded dd cols are zeros
    }
}

// ---------------------------------------------------------------------------
// K7: BN stats for merge output: per o over (b, dd<500)
// ---------------------------------------------------------------------------
__global__ void k7_bnstats_m(const float* __restrict__ mbuf,
                             float* __restrict__ mean, float* __restrict__ var) {
    const int o = blockIdx.x;
    const int tid = threadIdx.x;
    float s = 0.0f, s2 = 0.0f;
    for (int i = tid; i < BB * DD; i += blockDim.x) {
        int b = i / DD, dd = i % DD;
        float v = mbuf[((size_t)b * CO + o) * DP + dd];
        s += v; s2 += v * v;
    }
    __shared__ float ss[256], ss2[256];
    ss[tid] = s; ss2[tid] = s2;
    __syncthreads();
    for (int off = 128; off > 0; off >>= 1) {
        if (tid < off) { ss[tid] += ss[tid + off]; ss2[tid] += ss2[tid + off]; }
        __syncthreads();
    }
    if (tid == 0) {
        const float invN = 1.0f / (float)(BB * DD);
        float mu = ss[0] * invN;
        mean[o] = mu;
        var[o]  = ss2[0] * invN - mu * mu;
    }
}

// ---------------------------------------------------------------------------
// K8: out = relu(bn(m))  -> d_out [B, Co, 500]
// ---------------------------------------------------------------------------
__global__ void k8_final(const float* __restrict__ mbuf,
                         const float* __restrict__ mean, const float* __restrict__ var,
                         const float* __restrict__ g, const float* __restrict__ beta,
                         float* __restrict__ out) {
    size_t i = (size_t)blockIdx.x * blockDim.x + threadIdx.x;
    const size_t total = (size_t)BB * CO * DD;
    if (i >= total) return;
    int dd = (int)(i % DD);
    int o  = (int)((i / DD) % CO);
    int b  = (int)(i / ((size_t)DD * CO));
    float v = mbuf[((size_t)b * CO + o) * DP + dd];
    float t = (v - mean[o]) * rsqrtf(var[o] + BN_EPS) * g[o] + beta[o];
    out[i] = fmaxf(t, 0.0f);
}

extern "C" void kernel_launch(void* const* d_in, const int* in_sizes, int n_in,
                              void* d_out, int out_size, void* d_ws, size_t ws_size,
                              hipStream_t stream) {
    const float* x     = (const float*)d_in[0];
    const float* Wpool = (const float*)d_in[1];
    const float* Wpw   = (const float*)d_in[2];
    const float* g_pw  = (const float*)d_in[3];
    const float* b_pw  = (const float*)d_in[4];
    const float* Wm    = (const float*)d_in[5];
    const float* g_m   = (const float*)d_in[6];
    const float* b_m   = (const float*)d_in[7];
    float* out = (float*)d_out;

    float* ws = (float*)d_ws;
    float* rowmax  = ws;                                   // NH*BB*DP
    float* rowsum  = rowmax + (size_t)NH * BB * DP;        // NH*BB*DP
    float* ht      = rowsum + (size_t)NH * BB * DP;        // NH*BB*DP*CC
    float* ybuf    = ht + (size_t)NH * BB * DP * CC;       // NH*BB*DP*CC (y, then z in place)
    float* mean_pw = ybuf + (size_t)NH * BB * DP * CC;     // NH*DP
    float* var_pw  = mean_pw + (size_t)NH * DP;            // NH*DP
    float* mbuf    = var_pw + (size_t)NH * DP;             // BB*CO*DP
    float* mean_m  = mbuf + (size_t)BB * CO * DP;          // CO
    float* var_m   = mean_m + CO;                          // CO

    const int nbd = NH * BB * 32;                          // 2048 wave-sized WGs

    k1_softmax_stats<<<nbd, 32, 0, stream>>>(x, Wpool, rowmax, rowsum);
    k2_pool_apply  <<<nbd, 32, 0, stream>>>(x, Wpool, rowmax, rowsum, ht);
    k3_pw          <<<nbd, 32, 0, stream>>>(Wpw, ht, ybuf);
    k4_bnstats_pw  <<<NH * DP, 256, 0, stream>>>(ybuf, mean_pw, var_pw);
    {
        const size_t total = (size_t)NH * BB * DP * CC;
        k5_apply<<<(unsigned)((total + 255) / 256), 256, 0, stream>>>(
            ht, mean_pw, var_pw, g_pw, b_pw, ybuf);
    }
    k6_merge       <<<BB * 32, 256, 0, stream>>>(Wm, ybuf, mbuf);
    k7_bnstats_m   <<<CO, 256, 0, stream>>>(mbuf, mean_m, var_m);
    {
        const size_t total = (size_t)BB * CO * DD;
        k8_final<<<(unsigned)((total + 255) / 256), 256, 0, stream>>>(
            mbuf, mean_m, var_m, g_m, b_m, out);
    }
}